// GRU_16088947491196
// MI455X (gfx1250) — compile-verified
//
#include <hip/hip_runtime.h>
#include <hip/hip_bf16.h>
#include <math.h>
#include <stdint.h>

// ---- problem constants (from reference) ----
#define B_   64
#define L_   128
#define NN   31      // tree nodes
#define E_   128     // E == ENC == H
#define H_   128
#define G3   384     // 3*H
#define LBL_ 104
#define XLD  132     // padded LDS row stride for A tiles (132 % 64 = 4)
#define WLD  388     // padded LDS row stride for staged Whh^T (388 % 64 = 4)

typedef __attribute__((ext_vector_type(2))) float v2f;
typedef __attribute__((ext_vector_type(8))) float v8f;

__device__ __forceinline__ v8f wmma4(v2f a, v2f b, v8f c) {
  // D = A(16x4 f32) * B(4x16 f32) + C(16x16 f32)
  return __builtin_amdgcn_wmma_f32_16x16x4_f32(false, a, false, b, (short)0, c, false, false);
}

// async 16-byte global->LDS copy (tracked on ASYNCcnt)
__device__ __forceinline__ void async_g2l_b128(uint32_t lds_byte_addr, uint64_t gaddr) {
  asm volatile("global_load_async_to_lds_b128 %0, %1, off"
               :: "v"(lds_byte_addr), "v"(gaddr) : "memory");
}
__device__ __forceinline__ void wait_async0() {
  asm volatile("s_wait_asynccnt 0" ::: "memory");
}

// ---------------- weight transpose: dst[c*rows + r] = src[r*cols + c] ----------------
__global__ void transpose_k(const float* __restrict__ src, float* __restrict__ dst,
                            int rows, int cols) {
  int i = blockIdx.x * blockDim.x + threadIdx.x;
  if (i < rows * cols) {
    int r = i / cols, c = i % cols;
    dst[c * rows + r] = src[i];
  }
}

// ---------------- fused embed + tree-sum + projection + node-max ----------------
// one block (128 thr, 4 waves) per (b,l); GEMM: XS(32x128) @ WcT -> max over nodes
__global__ void __launch_bounds__(128) encode_k(const int* __restrict__ tokens,
                                                const float* __restrict__ emb,
                                                const float* __restrict__ WcT,   // (E,ENC) k-major
                                                const float* __restrict__ Wcb,
                                                float* __restrict__ encodes) {
  __shared__ __align__(16) float X[32 * XLD];
  const int bl  = blockIdx.x;
  const int tid = threadIdx.x;
  const int* tok = tokens + bl * NN;

  // async gather of 31 embedding rows (512B each) straight into LDS
  {
    const uint32_t xbase = (uint32_t)(uintptr_t)(&X[0]);
    for (int i = tid; i < NN * 32; i += 128) {      // 992 x 16B transfers
      int node = i >> 5, seg = i & 31;
      uint64_t g = (uint64_t)(uintptr_t)(emb + (long)tok[node] * E_) + (uint64_t)(seg * 16);
      uint32_t l = xbase + (uint32_t)(node * XLD * 4 + seg * 16);
      async_g2l_b128(l, g);
    }
    wait_async0();
  }
  __syncthreads();

  // bottom-up subtree sums (heap layout: children of p are 2p+1, 2p+2)
  for (int lvl = 3; lvl >= 0; --lvl) {
    int s = (1 << lvl) - 1, n = 1 << lvl;
    for (int idx = tid; idx < n * E_; idx += 128) {
      int p = s + (idx >> 7), e = idx & 127;
      X[p * XLD + e] += X[(2 * p + 1) * XLD + e] + X[(2 * p + 2) * XLD + e];
    }
    __syncthreads();
  }
  // pad row 31 = copy of leaf row 30 (never affects the max)
  X[31 * XLD + tid] = X[30 * XLD + tid];
  __syncthreads();

  const int wave = tid >> 5, lane = tid & 31;
  const int ln = lane & 15, kh = lane >> 4;
  const int nb = wave * 32;            // each wave owns 32 output columns

  v8f acc[2][2] = {{{}, {}}, {{}, {}}};
  for (int kb = 0; kb < E_; kb += 4) {
    int kk = kb + 2 * kh;
    v2f a0 = *(const v2f*)&X[(ln) * XLD + kk];
    v2f a1 = *(const v2f*)&X[(16 + ln) * XLD + kk];
    v2f b0, b1;
    b0.x = WcT[kk * E_ + nb + ln];        b0.y = WcT[(kk + 1) * E_ + nb + ln];
    b1.x = WcT[kk * E_ + nb + 16 + ln];   b1.y = WcT[(kk + 1) * E_ + nb + 16 + ln];
    acc[0][0] = wmma4(a0, b0, acc[0][0]);
    acc[0][1] = wmma4(a0, b1, acc[0][1]);
    acc[1][0] = wmma4(a1, b0, acc[1][0]);
    acc[1][1] = wmma4(a1, b1, acc[1][1]);
  }

  // add subtree-count * bias, max over the 31 real nodes (row31 duplicates row30)
  for (int nt = 0; nt < 2; ++nt) {
    int col = nb + nt * 16 + ln;
    float bias = Wcb[col];
    float cmax = -INFINITY;
    #pragma unroll
    for (int mt = 0; mt < 2; ++mt) {
      #pragma unroll
      for (int v = 0; v < 8; ++v) {
        int m = mt * 16 + v + 8 * kh;
        int lvlm = 31 - __clz(m + 1);
        float cnt = (m == 31) ? 1.f : (float)((1 << (5 - lvlm)) - 1);
        cmax = fmaxf(cmax, acc[mt][nt][v] + cnt * bias);
      }
    }
    cmax = fmaxf(cmax, __shfl_xor(cmax, 16, 32));   // combine m halves across lane groups
    if (kh == 0) encodes[bl * E_ + col] = cmax;
  }
}

// ---------------- gi = encodes @ Wih^T + bih  (M=8192, N=384, K=128) ----------------
// register-blocked: each wave computes a 32x48 job (2 M-tiles x 3 N-tiles),
// 6 wmma per k-step against 2 A-fragment + 3 B-fragment loads.
__global__ void __launch_bounds__(128) gi_k(const float* __restrict__ enc,
                                            const float* __restrict__ WT,   // (128,384) k-major
                                            const float* __restrict__ bias,
                                            float* __restrict__ gi) {
  const int wave = threadIdx.x >> 5, lane = threadIdx.x & 31;
  const int ln = lane & 15, kh = lane >> 4;
  const int job = blockIdx.x * 4 + wave;           // 2048 jobs total
  const int m0 = (job >> 3) * 32;                  // 256 M-pairs
  const int n0 = (job & 7) * 48;                   // 8 N-triples

  v8f acc[2][3] = {{{}, {}, {}}, {{}, {}, {}}};
  for (int kb = 0; kb < E_; kb += 4) {
    int kk = kb + 2 * kh;
    v2f a0 = *(const v2f*)&enc[(m0 + ln) * E_ + kk];
    v2f a1 = *(const v2f*)&enc[(m0 + 16 + ln) * E_ + kk];
    v2f b[3];
    #pragma unroll
    for (int nt = 0; nt < 3; ++nt) {
      b[nt].x = WT[kk * G3 + n0 + nt * 16 + ln];
      b[nt].y = WT[(kk + 1) * G3 + n0 + nt * 16 + ln];
    }
    #pragma unroll
    for (int nt = 0; nt < 3; ++nt) {
      acc[0][nt] = wmma4(a0, b[nt], acc[0][nt]);
      acc[1][nt] = wmma4(a1, b[nt], acc[1][nt]);
    }
  }
  #pragma unroll
  for (int nt = 0; nt < 3; ++nt) {
    float bb = bias[n0 + nt * 16 + ln];
    #pragma unroll
    for (int mt = 0; mt < 2; ++mt) {
      #pragma unroll
      for (int v = 0; v < 8; ++v) {
        int m = m0 + mt * 16 + v + 8 * kh;
        gi[(long)m * G3 + n0 + nt * 16 + ln] = acc[mt][nt][v] + bb;
      }
    }
  }
}

// ---------------- persistent GRU scan: 8 blocks = 2 dirs x 4 batch quarters ----------------
// dynamic LDS layout:
//   sWT  : 128 x WLD floats  (staged Whh^T, padded rows)    198,656 B
//   hbuf : 16 x XLD floats   (hidden state)                   8,448 B
//   gh   : 16 x G3 floats    (gate pre-activations)          24,576 B
__global__ void __launch_bounds__(256) scan_k(const float* __restrict__ gi_f,
                                              const float* __restrict__ gi_b,
                                              const float* __restrict__ WhhTf,
                                              const float* __restrict__ WhhTb,
                                              const float* __restrict__ bhhf,
                                              const float* __restrict__ bhhb,
                                              float* __restrict__ pooled) {
  extern __shared__ __align__(16) float smem[];
  float* sWT  = smem;                  // 128*WLD
  float* hbuf = sWT + 128 * WLD;       // 16*XLD
  float* gh   = hbuf + 16 * XLD;       // 16*G3

  const int dir = blockIdx.x >> 2;
  const int bq  = blockIdx.x & 3;                 // 16 batch rows per block
  const float* gi  = dir ? gi_b  : gi_f;
  const float* WT  = dir ? WhhTb : WhhTf;
  const float* bhh = dir ? bhhb  : bhhf;
  const int tid = threadIdx.x;
  const int wave = tid >> 5, lane = tid & 31;
  const int ln = lane & 15, kh = lane >> 4;

  // stage the whole Whh^T (128x384 f32) into LDS once, async, 16B granules
  {
    const uint64_t wg = (uint64_t)(uintptr_t)WT;
    const uint32_t wl = (uint32_t)(uintptr_t)sWT;
    for (int i = tid; i < 128 * 96; i += 256) {    // 96 x 16B per row
      int row = i / 96, seg = i % 96;
      async_g2l_b128(wl + (uint32_t)(row * (WLD * 4) + seg * 16),
                     wg + (uint64_t)(row * (G3 * 4) + seg * 16));
    }
    wait_async0();
  }
  for (int i = tid; i < 16 * XLD; i += 256) hbuf[i] = 0.f;
  float mx[8];
  #pragma unroll
  for (int j = 0; j < 8; ++j) mx[j] = -INFINITY;
  __syncthreads();

  for (int ti = 0; ti < L_; ++ti) {
    int t = dir ? (L_ - 1 - ti) : ti;
    // gh = h @ Whh^T + bhh : 8 waves x 3 n-tiles = 24 tiles (N=384), M=16, K=128
    for (int i = 0; i < 3; ++i) {
      int n0 = (wave * 3 + i) * 16;
      v8f acc = {};
      for (int kb = 0; kb < H_; kb += 4) {
        int kk = kb + 2 * kh;
        v2f a = *(const v2f*)&hbuf[ln * XLD + kk];
        v2f b;
        b.x = sWT[kk * WLD + n0 + ln];
        b.y = sWT[(kk + 1) * WLD + n0 + ln];
        acc = wmma4(a, b, acc);
      }
      float bb = bhh[n0 + ln];
      #pragma unroll
      for (int v = 0; v < 8; ++v)
        gh[(v + 8 * kh) * G3 + n0 + ln] = acc[v] + bb;
    }
    // prefetch next step's gi slab (16 rows x 384 f32 = 192 cache lines of 128B)
    {
      int tn = dir ? (t - 1) : (t + 1);
      if ((unsigned)tn < (unsigned)L_ && tid < 192) {
        int r = tid / 12, seg = tid % 12;
        __builtin_prefetch(&gi[((long)(bq * 16 + r) * L_ + tn) * G3 + seg * 32], 0, 0);
      }
    }
    __syncthreads();
    // gates + state update: 16x128 elements, 8 per thread (fixed mapping -> max in regs)
    #pragma unroll
    for (int j = 0; j < 8; ++j) {
      int p = tid + 256 * j;
      int r = p >> 7, c = p & 127;
      long gbase = ((long)(bq * 16 + r) * L_ + t) * G3;
      float ir = gi[gbase + c], iz = gi[gbase + 128 + c], in = gi[gbase + 256 + c];
      float hr = gh[r * G3 + c], hz = gh[r * G3 + 128 + c], hn = gh[r * G3 + 256 + c];
      float rg = 1.f / (1.f + __expf(-(ir + hr)));
      float z  = 1.f / (1.f + __expf(-(iz + hz)));
      float nn = tanhf(in + rg * hn);
      float hnew = (1.f - z) * nn + z * hbuf[r * XLD + c];
      hbuf[r * XLD + c] = hnew;
      mx[j] = fmaxf(mx[j], hnew);
    }
    __syncthreads();
  }
  #pragma unroll
  for (int j = 0; j < 8; ++j) {
    int p = tid + 256 * j;
    int r = p >> 7, c = p & 127;
    pooled[(bq * 16 + r) * 256 + dir * 128 + c] = mx[j];
  }
}

// ---------------- out = pooled @ Wout^T + bout ----------------
__global__ void out_k(const float* __restrict__ pooled,
                      const float* __restrict__ Wout,
                      const float* __restrict__ bout,
                      float* __restrict__ out) {
  int i = blockIdx.x * blockDim.x + threadIdx.x;
  if (i >= B_ * LBL_) return;
  int b = i / LBL_, o = i % LBL_;
  float s = bout[o];
  const float* p = pooled + b * 256;
  const float* w = Wout + o * 256;
  #pragma unroll 8
  for (int k = 0; k < 256; ++k) s += p[k] * w[k];
  out[i] = s;
}

extern "C" void kernel_launch(void* const* d_in, const int* in_sizes, int n_in,
                              void* d_out, int out_size, void* d_ws, size_t ws_size,
                              hipStream_t stream) {
  const int*   tokens = (const int*)  d_in[0];
  const float* emb    = (const float*)d_in[1];
  const float* Wc_w   = (const float*)d_in[2];
  const float* Wc_b   = (const float*)d_in[3];
  const float* Wih_f  = (const float*)d_in[4];
  const float* Whh_f  = (const float*)d_in[5];
  const float* bih_f  = (const float*)d_in[6];
  const float* bhh_f  = (const float*)d_in[7];
  const float* Wih_b  = (const float*)d_in[8];
  const float* Whh_b  = (const float*)d_in[9];
  const float* bih_b  = (const float*)d_in[10];
  const float* bhh_b  = (const float*)d_in[11];
  const float* Wout   = (const float*)d_in[12];
  const float* bout   = (const float*)d_in[13];

  float* ws     = (float*)d_ws;
  float* WcT    = ws;                    // 128*128
  float* WihTf  = WcT    + 16384;        // 128*384
  float* WhhTf  = WihTf  + 49152;
  float* WihTb  = WhhTf  + 49152;
  float* WhhTb  = WihTb  + 49152;
  float* enc    = WhhTb  + 49152;        // 8192*128
  float* gif    = enc    + 1048576;      // 8192*384
  float* gib    = gif    + 3145728;
  float* pooled = gib    + 3145728;      // 64*256
  float* out    = (float*)d_out;

  transpose_k<<<(128 * 128 + 255) / 256, 256, 0, stream>>>(Wc_w,  WcT,   128, 128);
  transpose_k<<<(384 * 128 + 255) / 256, 256, 0, stream>>>(Wih_f, WihTf, 384, 128);
  transpose_k<<<(384 * 128 + 255) / 256, 256, 0, stream>>>(Whh_f, WhhTf, 384, 128);
  transpose_k<<<(384 * 128 + 255) / 256, 256, 0, stream>>>(Wih_b, WihTb, 384, 128);
  transpose_k<<<(384 * 128 + 255) / 256, 256, 0, stream>>>(Whh_b, WhhTb, 384, 128);

  encode_k<<<B_ * L_, 128, 0, stream>>>(tokens, emb, WcT, Wc_b, enc);
  gi_k<<<512, 128, 0, stream>>>(enc, WihTf, bih_f, gif);
  gi_k<<<512, 128, 0, stream>>>(enc, WihTb, bih_b, gib);

  const size_t scan_lds = (size_t)(128 * WLD + 16 * XLD + 16 * G3) * sizeof(float);
  scan_k<<<8, 256, scan_lds, stream>>>(gif, gib, WhhTf, WhhTb, bhh_f, bhh_b, pooled);
  out_k<<<(B_ * LBL_ + 255) / 256, 256, 0, stream>>>(pooled, Wout, bout, out);
}